// KWinnerLayer2D_77464030151279
// MI455X (gfx1250) — compile-verified
//
#include <hip/hip_runtime.h>
#include <hip/hip_bf16.h>

// ---------------------------------------------------------------------------
// KWinner (per-batch top-k threshold) for MI455X / gfx1250.
// Radix-select over order-preserving uint keys, 11+11+10 bit passes with
// LDS-privatized histograms; all intermediate passes are L2-resident.
// Scan kernels stage their histogram slice via the CDNA5 async-to-LDS path
// (global_load_async_to_lds_b128 + s_wait_asynccnt).
// ---------------------------------------------------------------------------

#define BATCH      64
#define NPB        262144          // 256*32*32 elements per batch
#define K_TOP      13107u          // int(0.05 * 262144)
#define HIST_BINS  2048            // 11-bit histogram
#define DENSITY_F  0.05f
#define BOOST_F    1.5f

// Order-preserving float -> uint key (ascending float => ascending uint).
__device__ __forceinline__ unsigned fkey(float f) {
    unsigned u = __float_as_uint(f);
    return (u & 0x80000000u) ? ~u : (u | 0x80000000u);
}

// CDNA5 async copy: each lane copies 16 bytes from the global address gaddr to the LDS offset lds_off.
// VDST VGPR holds the 32-bit LDS byte offset (generic shared addr truncates
// to the LDS offset per ISA 10.2), VADDR holds the 64-bit global address.
__device__ __forceinline__ void async_g2lds_b128(unsigned lds_off, const void* gaddr) {
    asm volatile("global_load_async_to_lds_b128 %0, %1, off"
                 :: "v"(lds_off), "v"(gaddr) : "memory");
}
__device__ __forceinline__ void async_g2lds_b128_off16(unsigned lds_off, const void* gaddr) {
    asm volatile("global_load_async_to_lds_b128 %0, %1, off offset:16"
                 :: "v"(lds_off), "v"(gaddr) : "memory");
}
__device__ __forceinline__ void wait_asynccnt0() {
    asm volatile("s_wait_asynccnt 0x0" ::: "memory");
}

// ---- init: zero the global histograms and precompute the boost factor -----
__global__ __launch_bounds__(256) void init_k(const float* __restrict__ avg,
                                              float* __restrict__ factor,
                                              unsigned* __restrict__ hist) {
    int i = blockIdx.x * 256 + threadIdx.x;        // grid covers exactly NPB
    if (i < BATCH * HIST_BINS) hist[i] = 0u;
    factor[i] = expf((DENSITY_F - avg[i]) * BOOST_F);
}

// ---- histogram pass (LDS privatized, zero-skip flush) ---------------------
// PASS 0: bin = key >> 21                 (all elements)
// PASS 1: bin = (key >> 10) & 2047       (elements whose key>>21 == pref)
// PASS 2: bin = key & 1023               (elements whose key>>10 == pref)
template <int PASS>
__global__ __launch_bounds__(256) void hist_k(const float* __restrict__ x,
                                              const float* __restrict__ factor,
                                              unsigned* __restrict__ hist,
                                              const unsigned* __restrict__ prefArr) {
    __shared__ unsigned lh[HIST_BINS];
    const int tid = threadIdx.x;
    const int b   = blockIdx.y;

    for (int j = tid; j < HIST_BINS; j += 256) lh[j] = 0u;
    __syncthreads();

    const unsigned pref = (PASS == 0) ? 0u : prefArr[b];
    const float* xb = x + (size_t)b * NPB;
    const int base = blockIdx.x * 4096;            // 4096 elements per block

#pragma unroll
    for (int v = 0; v < 4; ++v) {
        const int i = base + (v * 256 + tid) * 4;
        // prefetch the next chunk of this batch's stream (global_prefetch_b8)
        __builtin_prefetch(xb + i + 4096, 0, 3);
        const float4 xv = *reinterpret_cast<const float4*>(xb + i);
        const float4 fv = *reinterpret_cast<const float4*>(factor + i);
        const float p[4] = {xv.x * fv.x, xv.y * fv.y, xv.z * fv.z, xv.w * fv.w};
#pragma unroll
        for (int e = 0; e < 4; ++e) {
            const unsigned key = fkey(p[e]);
            if (PASS == 0) {
                atomicAdd(&lh[key >> 21], 1u);                       // ds_add_u32
            } else if (PASS == 1) {
                if ((key >> 21) == pref) atomicAdd(&lh[(key >> 10) & 2047u], 1u);
            } else {
                if ((key >> 10) == pref) atomicAdd(&lh[key & 1023u], 1u);
            }
        }
    }
    __syncthreads();

    unsigned* gh = hist + b * HIST_BINS;
    for (int j = tid; j < HIST_BINS; j += 256) {
        const unsigned c = lh[j];
        if (c) atomicAdd(&gh[j], c);               // non-returning global atomic
    }
}

// ---- per-batch suffix scan: find bin where count-from-top crosses K -------
// Histogram slice is staged into LDS with async-to-LDS copies so both read
// phases (partial sums + crossing walk) come from LDS; the global slice is
// then re-zeroed for the next pass.
template <int PASS>
__global__ __launch_bounds__(256) void scan_k(unsigned* __restrict__ hist,
                                              unsigned* __restrict__ prefArr,
                                              unsigned* __restrict__ kArr,
                                              unsigned* __restrict__ threshArr) {
    const int NB  = (PASS == 2) ? 1024 : 2048;
    const int PER = NB / 256;                       // 8 or 4 bins per thread
    __shared__ __align__(16) unsigned lh[HIST_BINS];
    __shared__ unsigned partial[256];
    __shared__ unsigned above[256];

    const int b = blockIdx.x;
    const int t = threadIdx.x;
    unsigned* h = hist + b * HIST_BINS;
    const int lo = t * PER;

    // Stage this thread's bins: each lane copies exactly the bytes it will
    // read, so s_wait_asynccnt 0 alone orders the LDS writes for this wave.
    {
        const unsigned lds0 = (unsigned)(uintptr_t)(const void*)&lh[lo];
        async_g2lds_b128(lds0, h + lo);             // bins lo..lo+3
        if (PER == 8) async_g2lds_b128_off16(lds0, h + lo); // bins lo+4..lo+7
        wait_asynccnt0();
    }

    const unsigned K       = (PASS == 0) ? K_TOP : kArr[b];
    const unsigned oldPref = (PASS == 0) ? 0u : prefArr[b];

    unsigned s = 0;
    for (int j = 0; j < PER; ++j) s += lh[lo + j];
    partial[t] = s;
    __syncthreads();

    if (t == 0) {                                  // suffix sums over 256 partials
        unsigned a = 0;
        for (int i = 255; i >= 0; --i) { above[i] = a; a += partial[i]; }
    }
    __syncthreads();

    unsigned running = above[t];                   // count strictly above bin j
    for (int j = lo + PER - 1; j >= lo; --j) {
        const unsigned c = lh[j];
        if (running < K && running + c >= K) {     // exactly one (t, j) matches
            const unsigned q = (unsigned)j;
            if (PASS == 0)      { prefArr[b] = q;                     kArr[b] = K - running; }
            else if (PASS == 1) { prefArr[b] = (oldPref << 11) | q;   kArr[b] = K - running; }
            else                { threshArr[b] = (oldPref << 10) | q; }
        }
        running += c;
    }
    // re-zero the global slice for the next pass (reads all came from LDS)
    for (int j = 0; j < PER; ++j) h[lo + j] = 0u;
}

// ---- apply: out = xb if key(xb) >= threshKey[b] else 0 --------------------
__global__ __launch_bounds__(256) void apply_k(const float* __restrict__ x,
                                               const float* __restrict__ factor,
                                               const unsigned* __restrict__ threshArr,
                                               float* __restrict__ out) {
    const int b = blockIdx.y;
    const unsigned tk = threshArr[b];
    const int i = (blockIdx.x * 256 + threadIdx.x) * 4;
    const float* xb = x + (size_t)b * NPB;
    __builtin_prefetch(xb + i + 1024, 0, 3);

    const float4 xv = *reinterpret_cast<const float4*>(xb + i);
    const float4 fv = *reinterpret_cast<const float4*>(factor + i);
    float4 o;
    float p;
    p = xv.x * fv.x; o.x = (fkey(p) >= tk) ? p : 0.0f;
    p = xv.y * fv.y; o.y = (fkey(p) >= tk) ? p : 0.0f;
    p = xv.z * fv.z; o.z = (fkey(p) >= tk) ? p : 0.0f;
    p = xv.w * fv.w; o.w = (fkey(p) >= tk) ? p : 0.0f;
    *reinterpret_cast<float4*>(out + (size_t)b * NPB + i) = o;
}

// ---------------------------------------------------------------------------
extern "C" void kernel_launch(void* const* d_in, const int* in_sizes, int n_in,
                              void* d_out, int out_size, void* d_ws, size_t ws_size,
                              hipStream_t stream) {
    const float* x   = (const float*)d_in[0];   // (64, 256, 32, 32)
    const float* avg = (const float*)d_in[1];   // (256, 32, 32)
    float* out = (float*)d_out;

    // workspace layout (all re-initialized every call; graph-replay safe)
    char* ws = (char*)d_ws;
    unsigned* hist      = (unsigned*)ws;                         // 64*2048*4 = 512 KiB
    float*    factor    = (float*)(ws + 524288);                 // 1 MiB
    unsigned* prefArr   = (unsigned*)(ws + 524288 + 1048576);    // 64 u32
    unsigned* kArr      = prefArr + 64;                          // 64 u32
    unsigned* threshArr = prefArr + 128;                         // 64 u32

    // 1) factor + zeroed histograms
    init_k<<<NPB / 256, 256, 0, stream>>>(avg, factor, hist);

    const dim3 hg(64, BATCH);   // 64 chunks * 64 batches, 4096 elems/block
    // 2) radix select: 11 + 11 + 10 bits
    hist_k<0><<<hg, 256, 0, stream>>>(x, factor, hist, prefArr);
    scan_k<0><<<BATCH, 256, 0, stream>>>(hist, prefArr, kArr, threshArr);
    hist_k<1><<<hg, 256, 0, stream>>>(x, factor, hist, prefArr);
    scan_k<1><<<BATCH, 256, 0, stream>>>(hist, prefArr, kArr, threshArr);
    hist_k<2><<<hg, 256, 0, stream>>>(x, factor, hist, prefArr);
    scan_k<2><<<BATCH, 256, 0, stream>>>(hist, prefArr, kArr, threshArr);

    // 3) threshold + mask, fully vectorized (b128)
    apply_k<<<dim3(NPB / 1024, BATCH), 256, 0, stream>>>(x, factor, threshArr, out);
}